// NGCFConv_90890097918587
// MI455X (gfx1250) — compile-verified
//
#include <hip/hip_runtime.h>
#include <hip/hip_bf16.h>

typedef __attribute__((ext_vector_type(2))) float v2f;
typedef __attribute__((ext_vector_type(8))) float v8f;

#define NGCF_D 64

// ---------------------------------------------------------------------------
// Phase 1: degree of each dst node (segment_sum of ones over dst)
// ---------------------------------------------------------------------------
__global__ void ngcf_degree_kernel(const int* __restrict__ dst,
                                   float* __restrict__ deg, int E) {
    int e = blockIdx.x * blockDim.x + threadIdx.x;
    if (e < E) atomicAdd(&deg[dst[e]], 1.0f);
}

// ---------------------------------------------------------------------------
// Phase 2: inv_sqrt[i] = deg>0 ? rsqrt(deg) : 0
// ---------------------------------------------------------------------------
__global__ void ngcf_invsqrt_kernel(const float* __restrict__ deg,
                                    float* __restrict__ inv, int N) {
    int i = blockIdx.x * blockDim.x + threadIdx.x;
    if (i < N) {
        float d = deg[i];
        inv[i] = (d > 0.0f) ? rsqrtf(d) : 0.0f;
    }
}

// ---------------------------------------------------------------------------
// Phase 3: SpMM  h[dst] += x[src] * (inv[src]*inv[dst])
// 64 threads per edge (one per feature). x (16MB) and h (16MB) are L2-resident
// on MI455X (192MB L2), so the fp32 atomics resolve in L2, not HBM.
// ---------------------------------------------------------------------------
__global__ void ngcf_spmm_kernel(const float* __restrict__ x,
                                 const int* __restrict__ src,
                                 const int* __restrict__ dst,
                                 const float* __restrict__ inv,
                                 float* __restrict__ h, int E) {
    int t = blockIdx.x * blockDim.x + threadIdx.x;
    int e = t >> 6;          // edge id
    int d = t & 63;          // feature id
    if (e >= E) return;
    int s  = src[e];
    int dd = dst[e];
    float w = inv[s] * inv[dd];
    atomicAdd(&h[dd * NGCF_D + d], x[s * NGCF_D + d] * w);
}

// ---------------------------------------------------------------------------
// Phase 4: out = leaky(h@W1 + b1) + leaky((x*h)@W2 + b2)
// Full-fp32 WMMA path: v_wmma_f32_16x16x4_f32.
//   block  = 128 threads (4 waves), one 16-row node tile per block
//   wave w = 16-column N-tile n0 = 16*w, computes BOTH GEMMs for its tile
// A layout (16x4 f32): lane half selects K-pair (koff), v2f lanes hold K,K+1.
// B layout (4x16 f32): lane&15 = N column, {vgpr, lane half} = K within step.
// C layout (16x16 f32): c[v] holds M = v + 8*(lane>>4), N = n0 + (lane&15).
// ---------------------------------------------------------------------------
__global__ __launch_bounds__(128)
void ngcf_gemm_fused_kernel(const float* __restrict__ x,
                            const float* __restrict__ h,
                            const float* __restrict__ W1,
                            const float* __restrict__ b1,
                            const float* __restrict__ W2,
                            const float* __restrict__ b2,
                            float* __restrict__ out) {
    const int m0   = blockIdx.x * 16;
    const int wave = threadIdx.x >> 5;
    const int lane = threadIdx.x & 31;
    const int n0   = wave * 16;
    const int row  = lane & 15;      // M (A) / N (B,C) sub-index
    const int half = lane >> 4;
    const int koff = half * 2;       // K-pair selected by lane half
    const int col  = n0 + row;       // output column for B/C fragments

    const float* __restrict__ hrow = h + (size_t)(m0 + row) * NGCF_D;
    const float* __restrict__ xrow = x + (size_t)(m0 + row) * NGCF_D;

    v8f c1 = {};
    v8f c2 = {};

    #pragma unroll
    for (int k = 0; k < NGCF_D; k += 4) {
        // A fragments: contiguous float2 (K, K+1) at K = k + koff
        v2f a1 = *(const v2f*)(hrow + k + koff);          // h
        v2f xv = *(const v2f*)(xrow + k + koff);
        v2f a2 = a1 * xv;                                 // x*h elementwise

        // B fragments: W[K][col], W[K+1][col]  (K = k + koff)
        v2f bb1, bb2;
        bb1.x = W1[(k + koff + 0) * NGCF_D + col];
        bb1.y = W1[(k + koff + 1) * NGCF_D + col];
        bb2.x = W2[(k + koff + 0) * NGCF_D + col];
        bb2.y = W2[(k + koff + 1) * NGCF_D + col];

        // D = A x B + C, fp32 in/out
        c1 = __builtin_amdgcn_wmma_f32_16x16x4_f32(
                false, a1, false, bb1, (short)0, c1, false, false);
        c2 = __builtin_amdgcn_wmma_f32_16x16x4_f32(
                false, a2, false, bb2, (short)0, c2, false, false);
    }

    const float bias1 = b1[col];
    const float bias2 = b2[col];

    #pragma unroll
    for (int v = 0; v < 8; ++v) {
        int m = m0 + v + 8 * half;
        float o1 = c1[v] + bias1;
        o1 = (o1 >= 0.0f) ? o1 : 0.2f * o1;   // leaky_relu, alpha = 0.2
        float o2 = c2[v] + bias2;
        o2 = (o2 >= 0.0f) ? o2 : 0.2f * o2;
        out[(size_t)m * NGCF_D + col] = o1 + o2;
    }
}

// ---------------------------------------------------------------------------
// Launcher
// Inputs (setup_inputs order): x[N*64] f32, edge_index[2*E] int,
//                              W1[64*64], b1[64], W2[64*64], b2[64]
// Workspace: deg[N] | inv[N] | h[N*64]  (all f32)
// ---------------------------------------------------------------------------
extern "C" void kernel_launch(void* const* d_in, const int* in_sizes, int n_in,
                              void* d_out, int out_size, void* d_ws, size_t ws_size,
                              hipStream_t stream) {
    const float* x  = (const float*)d_in[0];
    const int*   ei = (const int*)d_in[1];
    const float* W1 = (const float*)d_in[2];
    const float* b1 = (const float*)d_in[3];
    const float* W2 = (const float*)d_in[4];
    const float* b2 = (const float*)d_in[5];
    float* out = (float*)d_out;

    const int N = in_sizes[0] / NGCF_D;   // 65536
    const int E = in_sizes[1] / 2;        // 1048576
    const int* src = ei;                  // edge_index[0, :]
    const int* dst = ei + E;              // edge_index[1, :]

    float* deg = (float*)d_ws;
    float* inv = deg + N;
    float* h   = inv + N;

    // zero deg + inv + h (h must be zero before scatter-add)
    hipMemsetAsync(d_ws, 0, (size_t)(2 * N + (size_t)N * NGCF_D) * sizeof(float),
                   stream);

    ngcf_degree_kernel<<<(E + 255) / 256, 256, 0, stream>>>(dst, deg, E);
    ngcf_invsqrt_kernel<<<(N + 255) / 256, 256, 0, stream>>>(deg, inv, N);

    long long spmm_threads = (long long)E * NGCF_D;
    int spmm_blocks = (int)((spmm_threads + 255) / 256);
    ngcf_spmm_kernel<<<spmm_blocks, 256, 0, stream>>>(x, src, dst, inv, h, E);

    ngcf_gemm_fused_kernel<<<N / 16, 128, 0, stream>>>(x, h, W1, b1, W2, b2, out);
}